// inference_model_25116968747267
// MI455X (gfx1250) — compile-verified
//
#include <hip/hip_runtime.h>
#include <cstdint>
#include <cstddef>

namespace {

constexpr int       kNPG = 1024;
constexpr int       kB   = 256;
constexpr int       kN   = kNPG * kB;     // 262144 nodes
constexpr long long kE   = 16LL * kN;     // 4194304 edges
constexpr int       kH   = 128;
constexpr int       kZ   = 64;
constexpr float     kEps = 1e-5f;

typedef float v2f __attribute__((ext_vector_type(2)));
typedef float v8f __attribute__((ext_vector_type(8)));

// ---------------------------------------------------------------------------
// Generic f32 WMMA GEMM:  out[z] = A[z] @ B[z] (+bias | +=out)  using
// V_WMMA_F32_16X16X4_F32.  A block covers 32 rows x (waves*16) cols; each
// wave computes TWO 16x16 tiles (row tiles 0/1, same col tile) so each B
// fragment feeds two WMMAs.  K is a template constant (multiple of 64) so the
// K-chunk loop fully unrolls -> 32 back-to-back v_wmma per chunk.
// Layout (ISA 7.12.2, 32-bit operands):
//   A frag: lane L, vgpr v -> A[L%16][k + v + 2*(L/16)]
//   B frag: lane L, vgpr v -> B[k + v + 2*(L/16)][L%16]
//   D frag: lane L, vgpr v -> D[v + 8*(L/16)][L%16]
// ---------------------------------------------------------------------------
template <int K>
__global__ __launch_bounds__(256) void gemm_wmma_f32(
    const float* __restrict__ A, const float* __restrict__ B,
    const float* __restrict__ bias, float* __restrict__ out,
    int lda, int ldb, int ldd,
    long long Abatch, long long Bbatch, long long Obatch, int accumulate)
{
  static_assert(K % 64 == 0 && K <= 128, "K must be 64 or 128");
  __shared__ float Asm[32 * 132];   // 32 x K tile (row-tile pair), padded
  __shared__ float Bsm[64 * 144];   // K-chunk (64) x up-to-128 cols, padded

  const long long z = blockIdx.z;
  A   += z * Abatch;
  B   += z * Bbatch;
  out += z * Obatch;

  const int tid  = threadIdx.x;
  const int nt   = blockDim.x;
  const int wave = tid >> 5;
  const int lane = tid & 31;
  const int half = lane >> 4;     // 0: lanes 0-15, 1: lanes 16-31
  const int l16  = lane & 15;

  const int row0  = blockIdx.x * 32;
  const int bcols = nt >> 1;      // 16 * waves_per_block
  const int col0  = wave * 16;

  // Stage A tile (32 x K).
  for (int idx = tid; idx < 32 * K; idx += nt) {
    int r = idx / K, c = idx - r * K;
    Asm[r * 132 + c] = A[(long long)(row0 + r) * lda + c];
  }

  v8f acc0, acc1;
  if (accumulate) {
#pragma unroll
    for (int v = 0; v < 8; ++v) {
      int m = v + 8 * half;
      acc0[v] = out[(long long)(row0 + m) * ldd + col0 + l16];
      acc1[v] = out[(long long)(row0 + 16 + m) * ldd + col0 + l16];
    }
  } else {
    float bv = bias ? bias[col0 + l16] : 0.0f;
#pragma unroll
    for (int v = 0; v < 8; ++v) { acc0[v] = bv; acc1[v] = bv; }
  }

  for (int kk = 0; kk < K; kk += 64) {
    __syncthreads();
    for (int idx = tid; idx < 64 * bcols; idx += nt) {
      int r = idx / bcols, c = idx - r * bcols;
      Bsm[r * 144 + c] = B[(long long)(kk + r) * ldb + c];
    }
    __syncthreads();
#pragma unroll
    for (int k = 0; k < 64; k += 4) {
      v2f a0, a1, b;
      const float* ap0 = &Asm[l16 * 132 + kk + k + 2 * half];
      const float* ap1 = ap0 + 16 * 132;
      a0.x = ap0[0]; a0.y = ap0[1];
      a1.x = ap1[0]; a1.y = ap1[1];
      b.x = Bsm[(k + 2 * half + 0) * 144 + col0 + l16];
      b.y = Bsm[(k + 2 * half + 1) * 144 + col0 + l16];
      acc0 = __builtin_amdgcn_wmma_f32_16x16x4_f32(false, a0, false, b,
                                                   (short)0, acc0, false, false);
      acc1 = __builtin_amdgcn_wmma_f32_16x16x4_f32(false, a1, false, b,
                                                   (short)0, acc1, false, false);
    }
  }

#pragma unroll
  for (int v = 0; v < 8; ++v) {
    int m = v + 8 * half;
    out[(long long)(row0 + m) * ldd + col0 + l16] = acc0[v];
    out[(long long)(row0 + 16 + m) * ldd + col0 + l16] = acc1[v];
  }
}

// ---------------------------------------------------------------------------
// Degree / normalization
// ---------------------------------------------------------------------------
__global__ void deg_init(float* __restrict__ deg) {
  long long i = (long long)blockIdx.x * blockDim.x + threadIdx.x;
  if (i < kN) deg[i] = 1.0f;   // self-loop
}

__global__ void deg_scatter(const int* __restrict__ ei, float* __restrict__ deg) {
  long long e = (long long)blockIdx.x * blockDim.x + threadIdx.x;
  if (e < kE) unsafeAtomicAdd(&deg[ei[kE + e]], 1.0f);
}

__global__ void deg_finalize(float* __restrict__ deg) {
  long long i = (long long)blockIdx.x * blockDim.x + threadIdx.x;
  if (i < kN) deg[i] = rsqrtf(deg[i]);
}

// ---------------------------------------------------------------------------
// GCN aggregation: self-loop init then edge atomic scatter
// ---------------------------------------------------------------------------
__global__ void init_agg(const float* __restrict__ xw, const float* __restrict__ dinv,
                         const float* __restrict__ b, float* __restrict__ agg)
{
  long long idx = (long long)blockIdx.x * blockDim.x + threadIdx.x;
  long long n = idx >> 7;
  int c = (int)(idx & (kH - 1));
  float di = dinv[n];
  agg[idx] = xw[idx] * di * di + b[c];
}

__global__ void edge_scatter(const int* __restrict__ ei, const float* __restrict__ xw,
                             const float* __restrict__ dinv, float* __restrict__ agg)
{
  long long t = (long long)blockIdx.x * blockDim.x + threadIdx.x;  // kE * 128 threads
  long long e = t >> 7;
  int c = (int)(t & (kH - 1));
  int s = ei[e];
  int d = ei[kE + e];
  float nrm = dinv[s] * dinv[d];
  unsafeAtomicAdd(&agg[(long long)d * kH + c], xw[(long long)s * kH + c] * nrm);
}

// ---------------------------------------------------------------------------
// BatchNorm (training-mode biased stats) + ReLU
// ---------------------------------------------------------------------------
__global__ void zero_stats(float* __restrict__ stats) {
  stats[threadIdx.x] = 0.0f;   // 256 entries: [0..127]=sum, [128..255]=sumsq
}

__global__ __launch_bounds__(256) void bn_stats(const float* __restrict__ hbuf,
                                                float* __restrict__ stats)
{
  __shared__ float sh0[256];
  __shared__ float sh1[256];
  const int t = threadIdx.x;
  const int c = t & (kH - 1);
  const int hh = t >> 7;                    // 0 or 1
  const long long r0 = (long long)blockIdx.x * 512;
  float s = 0.f, ss = 0.f;
  for (int r = hh; r < 512; r += 2) {
    float v = hbuf[(r0 + r) * kH + c];
    s += v; ss += v * v;
  }
  sh0[t] = s; sh1[t] = ss;
  __syncthreads();
  if (t < kH) {
    unsafeAtomicAdd(&stats[c],      sh0[c] + sh0[c + 128]);
    unsafeAtomicAdd(&stats[kH + c], sh1[c] + sh1[c + 128]);
  }
}

__global__ void bn_apply_relu(float* __restrict__ hbuf, const float* __restrict__ stats,
                              const float* __restrict__ g, const float* __restrict__ be)
{
  long long idx = (long long)blockIdx.x * blockDim.x + threadIdx.x;
  int c = (int)(idx & (kH - 1));
  float mu  = stats[c] * (1.0f / kN);
  float var = stats[kH + c] * (1.0f / kN) - mu * mu;
  float v = (hbuf[idx] - mu) * rsqrtf(var + kEps) * g[c] + be[c];
  hbuf[idx] = v > 0.f ? v : 0.f;
}

// ---------------------------------------------------------------------------
// phi = h @ w (+ b), one wave per node
// ---------------------------------------------------------------------------
__global__ __launch_bounds__(256) void phi_kernel(const float* __restrict__ h,
                                                  const float* __restrict__ wv,
                                                  const float* __restrict__ bptr,
                                                  float* __restrict__ phi, int add)
{
  int wave = threadIdx.x >> 5, lane = threadIdx.x & 31;
  long long n = (long long)blockIdx.x * 8 + wave;
  const float* row = h + n * kH;
  float acc = 0.f;
  for (int c = lane; c < kH; c += 32) acc += row[c] * wv[c];
  for (int off = 16; off > 0; off >>= 1) acc += __shfl_down(acc, off, 32);
  if (lane == 0) {
    float r = acc + (bptr ? bptr[0] : 0.f);
    phi[n] = add ? (phi[n] + r) : r;
  }
}

// ---------------------------------------------------------------------------
// Per-graph softmax attention + weighted pooling of h3 (graphs are contiguous)
// ---------------------------------------------------------------------------
__global__ __launch_bounds__(256) void graph_pool(const float* __restrict__ phi,
                                                  const float* __restrict__ h3,
                                                  float* __restrict__ hg)
{
  __shared__ float sh[256];
  __shared__ float m_s, z_s;
  const int g = blockIdx.x;
  const int t = threadIdx.x;
  const long long base = (long long)g * kNPG;

  float lm = -3.4e38f;
  for (int i = t; i < kNPG; i += 256) lm = fmaxf(lm, phi[base + i]);
  sh[t] = lm; __syncthreads();
  for (int off = 128; off > 0; off >>= 1) {
    if (t < off) sh[t] = fmaxf(sh[t], sh[t + off]);
    __syncthreads();
  }
  if (t == 0) m_s = sh[0];
  __syncthreads();
  const float m = m_s;

  float lz = 0.f;
  for (int i = t; i < kNPG; i += 256) lz += expf(phi[base + i] - m);
  sh[t] = lz; __syncthreads();
  for (int off = 128; off > 0; off >>= 1) {
    if (t < off) sh[t] += sh[t + off];
    __syncthreads();
  }
  if (t == 0) z_s = sh[0];
  __syncthreads();
  const float zinv = 1.0f / z_s;

  const int c = t & (kH - 1);
  const int hh = t >> 7;
  float acc = 0.f;
  for (int i = hh * 512; i < hh * 512 + 512; ++i) {
    long long n = base + i;
    acc += h3[n * kH + c] * expf(phi[n] - m);
  }
  __syncthreads();
  sh[t] = acc; __syncthreads();
  if (t < kH) hg[(long long)g * kH + t] = (sh[t] + sh[t + 128]) * zinv;
}

// ---------------------------------------------------------------------------
// z = mu + eps * exp(0.5 * logvar)
// ---------------------------------------------------------------------------
__global__ void reparam(const float* __restrict__ mu, const float* __restrict__ lv,
                        const float* __restrict__ eps, float* __restrict__ zo,
                        long long count)
{
  long long i = (long long)blockIdx.x * blockDim.x + threadIdx.x;
  if (i < count) zo[i] = mu[i] + eps[i] * expf(0.5f * lv[i]);
}

}  // namespace

extern "C" void kernel_launch(void* const* d_in, const int* in_sizes, int n_in,
                              void* d_out, int out_size, void* d_ws, size_t ws_size,
                              hipStream_t stream)
{
  (void)in_sizes; (void)n_in; (void)out_size; (void)ws_size;

  const float* x     = (const float*)d_in[0];
  const float* epsA  = (const float*)d_in[1];
  const float* epsi  = (const float*)d_in[2];
  const float* W1    = (const float*)d_in[3];  const float* b1  = (const float*)d_in[4];
  const float* W2    = (const float*)d_in[5];  const float* b2  = (const float*)d_in[6];
  const float* W3    = (const float*)d_in[7];  const float* b3  = (const float*)d_in[8];
  const float* g1    = (const float*)d_in[9];  const float* be1 = (const float*)d_in[10];
  const float* g2    = (const float*)d_in[11]; const float* be2 = (const float*)d_in[12];
  const float* g3    = (const float*)d_in[13]; const float* be3 = (const float*)d_in[14];
  const float* phi1w = (const float*)d_in[15];
  const float* phi1b = (const float*)d_in[16];
  const float* phi2w = (const float*)d_in[17];
  const float* muAw  = (const float*)d_in[18]; const float* muAb = (const float*)d_in[19];
  const float* lvAw  = (const float*)d_in[20]; const float* lvAb = (const float*)d_in[21];
  const float* Whw   = (const float*)d_in[22]; const float* Whb  = (const float*)d_in[23];
  const float* WA    = (const float*)d_in[24];
  const float* muiw  = (const float*)d_in[25]; const float* muib = (const float*)d_in[26];
  const float* lviw  = (const float*)d_in[27]; const float* lvib = (const float*)d_in[28];
  const int*   ei    = (const int*)d_in[29];
  // d_in[30] = batch: contiguous (n / NPG), exploited analytically.

  float* outp = (float*)d_out;
  float* z_i  = outp;
  float* mu_i = outp + (long long)kN * kZ;
  float* lv_i = outp + 2LL * kN * kZ;
  float* z_A  = outp + 3LL * kN * kZ;
  float* mu_A = z_A + (long long)kB * kZ;
  float* lv_A = mu_A + (long long)kB * kZ;

  float* w = (float*)d_ws;
  float* dinv  = w;  w += kN;
  float* xw    = w;  w += (long long)kN * kH;
  float* bufB  = w;  w += (long long)kN * kH;   // h1, later reused for h3
  float* bufC  = w;  w += (long long)kN * kH;   // h2
  float* sbuf  = w;  w += (long long)kN * kZ;
  float* phi   = w;  w += kN;
  float* hg    = w;  w += (long long)kB * kH;
  float* stats = w;  w += 256;

  const int elemBlocks = (int)(((long long)kN * kH) / 256);  // 131072
  const int edgeBlocks = (int)((kE * kH) / 256);             // 2097152

  // Degree / D^-1/2
  deg_init    <<<kN / 256, 256, 0, stream>>>(dinv);
  deg_scatter <<<(int)(kE / 256), 256, 0, stream>>>(ei, dinv);
  deg_finalize<<<kN / 256, 256, 0, stream>>>(dinv);

  auto layer = [&](const float* in, float* outb, const float* W, const float* b,
                   const float* g, const float* be) {
    gemm_wmma_f32<128><<<dim3(kN / 32, 1, 1), 256, 0, stream>>>(
        in, W, nullptr, xw, kH, kH, kH, 0, 0, 0, 0);
    init_agg     <<<elemBlocks, 256, 0, stream>>>(xw, dinv, b, outb);
    edge_scatter <<<edgeBlocks, 256, 0, stream>>>(ei, xw, dinv, outb);
    zero_stats   <<<1, 256, 0, stream>>>(stats);
    bn_stats     <<<kN / 512, 256, 0, stream>>>(outb, stats);
    bn_apply_relu<<<elemBlocks, 256, 0, stream>>>(outb, stats, g, be);
  };

  // h1 (bufB)
  layer(x, bufB, W1, b1, g1, be1);
  phi_kernel<<<kN / 8, 256, 0, stream>>>(bufB, phi1w, phi1b, phi, 0);
  // h2 (bufC)
  layer(bufB, bufC, W2, b2, g2, be2);
  phi_kernel<<<kN / 8, 256, 0, stream>>>(bufC, phi2w, nullptr, phi, 1);
  // h3 (reuse bufB)
  layer(bufC, bufB, W3, b3, g3, be3);

  // Attention pool -> hg [B, H]
  graph_pool<<<kB, 256, 0, stream>>>(phi, bufB, hg);

  // Graph-level heads: mu_A / logvar_A / z_A
  gemm_wmma_f32<128><<<dim3(kB / 32, 1, 1), 128, 0, stream>>>(
      hg, muAw, muAb, mu_A, kH, kZ, kZ, 0, 0, 0, 0);
  gemm_wmma_f32<128><<<dim3(kB / 32, 1, 1), 128, 0, stream>>>(
      hg, lvAw, lvAb, lv_A, kH, kZ, kZ, 0, 0, 0, 0);
  reparam<<<(kB * kZ) / 256, 256, 0, stream>>>(mu_A, lv_A, epsA, z_A, (long long)kB * kZ);

  // s = h3 @ Wh_w + Wh_b          [N, ZI]
  gemm_wmma_f32<128><<<dim3(kN / 32, 1, 1), 128, 0, stream>>>(
      bufB, Whw, Whb, sbuf, kH, kZ, kZ, 0, 0, 0, 0);

  // s += z_A @ WA[p]  for each node position p (batched over blockIdx.z):
  //   out offset = p*64 + g*65536 + col  == ((g*1024 + p)*64 + col)
  gemm_wmma_f32<64><<<dim3(kB / 32, 1, kNPG), 128, 0, stream>>>(
      z_A, WA, nullptr, sbuf, kZ, kZ, kZ * kNPG,
      0, (long long)kZ * kZ, (long long)kZ, 1);

  // mu_i / logvar_i / z_i
  gemm_wmma_f32<64><<<dim3(kN / 32, 1, 1), 128, 0, stream>>>(
      sbuf, muiw, muib, mu_i, kZ, kZ, kZ, 0, 0, 0, 0);
  gemm_wmma_f32<64><<<dim3(kN / 32, 1, 1), 128, 0, stream>>>(
      sbuf, lviw, lvib, lv_i, kZ, kZ, kZ, 0, 0, 0, 0);
  reparam<<<(int)(((long long)kN * kZ) / 256), 256, 0, stream>>>(
      mu_i, lv_i, epsi, z_i, (long long)kN * kZ);
}